// TimeNeighborhoodAttention_68719476736022
// MI455X (gfx1250) — compile-verified
//
#include <hip/hip_runtime.h>
#include <hip/hip_bf16.h>

// Problem constants (match the reference)
#define KS      7
#define DILX    2
#define NHEADS  4
#define NFRAMES 3
#define CDIM    128
#define HD      32          // head dim = 128/4
#define HH      64
#define WW      64
#define NPIX    4096        // 64*64
#define NB      2
#define KTOT    (NFRAMES*CDIM)  // 384
#define KSTEPS  (KTOT/4)        // 96 WMMA K-steps (even)
#define NFRAG   (KSTEPS*8*32)   // 24576 v2f fragments of wz_w in LDS (196,608 B)

typedef __attribute__((ext_vector_type(2))) float v2f;
typedef __attribute__((ext_vector_type(8))) float v8f;

// NATTEN shifted window start (L=64, k=7, d=2):
//   ni = i-6;  ni<0 -> i%2;  i+6>=64 -> 50 + i%2 (since 64%2==0);  else ni
__device__ __forceinline__ int win_start(int i) {
    int ni = i - (KS / 2) * DILX;                   // i - 6
    if (ni < 0) return i % DILX;
    if (i + (KS / 2) * DILX >= HH) return HH - KS * DILX + (i % DILX); // 50 + i%2
    return ni;
}

// ---------------------------------------------------------------------------
// Kernel 1: dilated neighborhood attention, one thread per (b, head, frame,
// pixel). Online softmax (running max / sum / rescaled accumulator) so we
// never materialize the 49 logits. All loads are lane-coalesced across j.
// q/k/v live in L2 (28 MB < 192 MB), so the 49x K/V reuse is cache-served.
// ---------------------------------------------------------------------------
__global__ void natten_attn_kernel(const float* __restrict__ q,
                                   const float* __restrict__ kk,
                                   const float* __restrict__ vv,
                                   const float* __restrict__ rpb,
                                   float* __restrict__ xn)
{
    int tid = blockIdx.x * blockDim.x + threadIdx.x;  // exact fit: 384*256
    int j = tid & 63;
    int t = tid >> 6;
    int i = t & 63;
    t >>= 6;
    int f = t % NFRAMES;
    t /= NFRAMES;
    int head = t & 3;
    int b = t >> 2;

    int si = win_start(i), sj = win_start(j);
    // (si - i) is always an exact multiple of DILX -> C division is exact
    int pbi = (si - i) / DILX + (KS - 1);
    int pbj = (sj - j) / DILX + (KS - 1);

    const float scale = 0.17677669529663687f;   // 32^-0.5

    const float* qp = q + ((size_t)(b * CDIM + head * HD)) * NPIX + i * WW + j;
    float qr[HD];
#pragma unroll
    for (int d = 0; d < HD; ++d) qr[d] = qp[(size_t)d * NPIX] * scale;

    const size_t kvbase = ((size_t)(b * KTOT + f * CDIM + head * HD)) * NPIX;
    const float* kb = kk + kvbase;
    const float* vb = vv + kvbase;
    const float* rp = rpb + head * (2 * KS - 1) * (2 * KS - 1);

    float m = -3.0e38f, s = 0.0f;
    float acc[HD];
#pragma unroll
    for (int d = 0; d < HD; ++d) acc[d] = 0.0f;

    for (int ti = 0; ti < KS; ++ti) {
        int ii  = si + ti * DILX;
        int bro = (pbi + ti) * (2 * KS - 1);
        for (int tj = 0; tj < KS; ++tj) {
            int jj  = sj + tj * DILX;
            int off = ii * WW + jj;

            float l = rp[bro + pbj + tj];
#pragma unroll
            for (int d = 0; d < HD; ++d)
                l += qr[d] * kb[(size_t)d * NPIX + off];

            float mn   = fmaxf(m, l);
            float corr = __expf(m - mn);   // 0 on the first tap (m = -3e38)
            float w    = __expf(l - mn);
            s = s * corr + w;
            m = mn;
#pragma unroll
            for (int d = 0; d < HD; ++d)
                acc[d] = acc[d] * corr + w * vb[(size_t)d * NPIX + off];
        }
    }

    float inv = 1.0f / s;
    // xn channel layout matches the reference concat: c = f*128 + head*32 + d
    float* op = xn + ((size_t)(b * KTOT + f * CDIM + head * HD)) * NPIX + i * WW + j;
#pragma unroll
    for (int d = 0; d < HD; ++d) op[(size_t)d * NPIX] = acc[d] * inv;
}

// ---------------------------------------------------------------------------
// Kernel 2: 1x1 conv as GEMM  out[b,o,p] = sum_c W[o,c]*xn[b,c,p] + bias[o]
// via V_WMMA_F32_16X16X4_F32 (full fp32 precision on the tensor pipe).
//
// - wz_w is staged once per block into LDS, PRE-PERMUTED into per-lane WMMA
//   A-fragment order: frag[(kstep*8+mt)*32 + lane] = {W[row][c], W[row][c+1]}
//   with row = mt*16+lane%16, c = kstep*4+2*(lane/16). Reads back are
//   conflict-free, 8B-aligned ds_load_b64 (the compiler fuses them into
//   ds_load_2addr_b64 pairs).
// - Wave tile = 4 M-tiles x 2 N-tiles: 8 v8f accumulators (64 VGPRs), no
//   scratch spills (round 2 spilled with 16 accumulators).
// - K-loop unrolled by 2 with two NAMED buffer sets (A/B) and no buffer
//   rotation: round 3's asm showed ~64 v_dual_mov register copies per
//   iteration resolving the acur<-anxt phis; alternating buffers kills them.
//   Each half-iteration prefetches the other buffer before its 8 WMMAs.
// - __launch_bounds__(256,1) gives the allocator full VGPR headroom.
//
// Fragment layouts (ISA 7.12.2, wave32):
//   A 16x4 : lane L holds M=L%16; {VGPR0,VGPR1} = K{0,1} (L<16) / K{2,3}
//   B 4x16 : lane L holds N=L%16; {VGPR0,VGPR1} = K{0,1} (L<16) / K{2,3}
//   C/D    : lane L holds N=L%16; VGPR v holds M = v + 8*(L>=16)
// ---------------------------------------------------------------------------
__global__ __launch_bounds__(256, 1)
void conv1x1_wmma_kernel(const float* __restrict__ xn,
                         const float* __restrict__ wzw,
                         const float* __restrict__ wzb,
                         float* __restrict__ out)
{
    extern __shared__ v2f wfrag[];   // NFRAG entries = 196,608 B

    // ---- cooperative staged+permuted load of wz_w into LDS ----
    for (int s = threadIdx.x; s < NFRAG; s += 256) {
        int ln    = s & 31;
        int mt    = (s >> 5) & 7;
        int kstep = s >> 8;
        int row   = mt * 16 + (ln & 15);
        int c     = kstep * 4 + 2 * (ln >> 4);
        const float* wp = wzw + (size_t)row * KTOT + c;   // contiguous pair
        v2f t;
        t.x = wp[0];
        t.y = wp[1];
        wfrag[s] = t;
    }
    __syncthreads();

    int lane  = threadIdx.x & 31;
    int wave  = threadIdx.x >> 5;
    int mhalf = wave & 1;                      // M-tiles mhalf*4 .. mhalf*4+3
    int ntp   = blockIdx.x * 4 + (wave >> 1);  // N-tile pair, 0..255
    int col   = lane & 15;      // N (pixel) within tile
    int hi    = lane >> 4;      // lane half selects K pair {0,1} vs {2,3}

    // two consecutive 16-pixel N-tiles per pair; 32 | 4096 so same batch b
    int g0 = ntp * 32 + col;
    int b  = g0 >> 12;
    int p0 = g0 & 4095;
    const float* xp0 = xn + (size_t)b * KTOT * NPIX + p0;
    const float* xp1 = xp0 + 16;               // N-tile 1 = +16 pixels

    v8f acc0[4], acc1[4];
#pragma unroll
    for (int mt = 0; mt < 4; ++mt)
#pragma unroll
        for (int e = 0; e < 8; ++e) { acc0[mt][e] = 0.0f; acc1[mt][e] = 0.0f; }

    // fragment loader for one K-step (this wave's 4 M-tiles only)
    auto load_step = [&](int ks, v2f a[4], v2f& b0, v2f& b1) {
        int c0 = ks * 4 + 2 * hi;
        b0.x = xp0[(size_t)c0 * NPIX];
        b0.y = xp0[(size_t)(c0 + 1) * NPIX];
        b1.x = xp1[(size_t)c0 * NPIX];
        b1.y = xp1[(size_t)(c0 + 1) * NPIX];
        const v2f* wf = wfrag + (size_t)ks * 256 + (mhalf * 4) * 32 + lane;
#pragma unroll
        for (int mt = 0; mt < 4; ++mt) a[mt] = wf[mt * 32];
    };

    v2f aA[4], bA0, bA1;       // buffer set A
    v2f aB[4], bB0, bB1;       // buffer set B
    load_step(0, aA, bA0, bA1);

    for (int ks = 0; ks < KSTEPS; ks += 2) {
        // prefetch ks+1 into B (always in range: KSTEPS even)
        load_step(ks + 1, aB, bB0, bB1);

#pragma unroll
        for (int mt = 0; mt < 4; ++mt) {
            acc0[mt] = __builtin_amdgcn_wmma_f32_16x16x4_f32(
                false, aA[mt], false, bA0, (short)0, acc0[mt], false, false);
            acc1[mt] = __builtin_amdgcn_wmma_f32_16x16x4_f32(
                false, aA[mt], false, bA1, (short)0, acc1[mt], false, false);
        }

        // prefetch ks+2 into A (clamp on the final tail; redundant load only)
        int ksn = (ks + 2 < KSTEPS) ? (ks + 2) : (KSTEPS - 1);
        load_step(ksn, aA, bA0, bA1);

#pragma unroll
        for (int mt = 0; mt < 4; ++mt) {
            acc0[mt] = __builtin_amdgcn_wmma_f32_16x16x4_f32(
                false, aB[mt], false, bB0, (short)0, acc0[mt], false, false);
            acc1[mt] = __builtin_amdgcn_wmma_f32_16x16x4_f32(
                false, aB[mt], false, bB1, (short)0, acc1[mt], false, false);
        }
    }

    // writeback: lane holds N=col, rows M = (mhalf*4+mt)*16 + e + 8*hi
    float* op0 = out + (size_t)b * CDIM * NPIX + p0;
    float* op1 = op0 + 16;
#pragma unroll
    for (int mt = 0; mt < 4; ++mt) {
#pragma unroll
        for (int e = 0; e < 8; ++e) {
            int mrow = (mhalf * 4 + mt) * 16 + e + 8 * hi;
            float bias = wzb[mrow];
            op0[(size_t)mrow * NPIX] = acc0[mt][e] + bias;
            op1[(size_t)mrow * NPIX] = acc1[mt][e] + bias;
        }
    }
}

extern "C" void kernel_launch(void* const* d_in, const int* in_sizes, int n_in,
                              void* d_out, int out_size, void* d_ws, size_t ws_size,
                              hipStream_t stream) {
    const float* q   = (const float*)d_in[0];
    const float* k   = (const float*)d_in[1];
    const float* v   = (const float*)d_in[2];
    const float* rpb = (const float*)d_in[3];
    const float* wzw = (const float*)d_in[4];
    const float* wzb = (const float*)d_in[5];
    float* out = (float*)d_out;
    float* xn  = (float*)d_ws;   // 2*384*4096 floats = 12.6 MB scratch

    // Attention: 2*4*3*4096 = 98304 threads, exact grid fit.
    natten_attn_kernel<<<384, 256, 0, stream>>>(q, k, v, rpb, xn);

    // GEMM: 512 waves = 64 blocks x 8 waves
    //   (even/odd waves take M-tiles 0-3 / 4-7; wave>>1 picks the N-tile pair)
    // 196,608 B dynamic LDS per block (WGP has 320 KB) — opt in explicitly.
    size_t lds_bytes = (size_t)NFRAG * sizeof(v2f);
    (void)hipFuncSetAttribute(reinterpret_cast<const void*>(conv1x1_wmma_kernel),
                              hipFuncAttributeMaxDynamicSharedMemorySize,
                              (int)lds_bytes);
    conv1x1_wmma_kernel<<<64, 256, lds_bytes, stream>>>(xn, wzw, wzb, out);
}